// TemporalFeatureFusion_35184372089080
// MI455X (gfx1250) — compile-verified
//
#include <hip/hip_runtime.h>

// ---------------------------------------------------------------------------
// MI455X (gfx1250). All GEMM-like math on the fp32 WMMA pipe
// (v_wmma_f32_16x16x4_f32): ~0.5 TFLOP fp32 vs a 67MB HBM stream at 23.3TB/s
// means the kernel is memory-bound, so fp32 matrix ops give reference
// precision at no roofline cost. Weight tiles are staged with the Tensor Data
// Mover (tensor_load_to_lds, TENSORcnt), contiguous activation rows with
// global_load_async_to_lds_b128 (ASYNCcnt), both double-buffered so DMA
// overlaps the WMMA loop.
// ---------------------------------------------------------------------------

typedef float v2f __attribute__((ext_vector_type(2)));
typedef float v8f __attribute__((ext_vector_type(8)));
typedef unsigned u32x4 __attribute__((ext_vector_type(4)));
typedef unsigned u32x8 __attribute__((ext_vector_type(8)));

#define DEVFN static __device__ __forceinline__

constexpr int B_ = 4, F_ = 16, C_ = 256, HW = 1024;
constexpr int Hd_ = 128;
constexpr float EPSV = 1e-5f;
constexpr int LDA = 36;  // A-tile LDS row stride (dwords): 16B aligned, bank-spread

// workspace layout (float offsets)
constexpr size_t OFF_XP   = 0;                                  // N*F*C
constexpr size_t OFF_ATTN = OFF_XP + (size_t)B_ * HW * F_ * C_; // B*C*HW
constexpr size_t OFF_H    = OFF_ATTN + (size_t)B_ * C_ * HW;
constexpr size_t OFF_C    = OFF_H + (size_t)B_ * Hd_ * HW;
constexpr size_t OFF_ACC  = OFF_C + (size_t)B_ * Hd_ * HW;
constexpr size_t OFF_C3A  = OFF_ACC + (size_t)B_ * Hd_ * HW;
constexpr size_t OFF_S3   = OFF_C3A + (size_t)B_ * Hd_ * HW;    // 512 floats
constexpr size_t OFF_S2   = OFF_S3 + 512;                       // 1024 floats
constexpr size_t OFF_CC   = OFF_S2 + 1024;                      // B*512*HW

DEVFN v8f wmma_f32(v2f a, v2f b, v8f c) {
  return __builtin_amdgcn_wmma_f32_16x16x4_f32(false, a, false, b, (short)0, c,
                                               false, false);
}

// A-matrix 16x4 fragment: lane = M row; half-wave selects K pair (ISA 7.12.2)
DEVFN v2f frag_a(const float* A, int lda, int lane, int k0) {
  const float* p = A + (size_t)(lane & 15) * lda + k0 + ((lane >> 4) << 1);
  v2f r; r.x = p[0]; r.y = p[1]; return r;
}
// B-matrix 4x16 fragment: lane = N col; VGPR pair = consecutive K rows
DEVFN v2f frag_b(const float* B, int ldb, int lane, int k0) {
  const float* p = B + (size_t)(k0 + ((lane >> 4) << 1)) * ldb + (lane & 15);
  v2f r; r.x = p[0]; r.y = p[ldb]; return r;
}

// Per-lane async copy of 16 contiguous bytes global -> LDS (ASYNCcnt).
DEVFN void async_b128(float* lds, const float* g) {
  unsigned lo = (unsigned)(size_t)lds;                 // low 32 bits = LDS offset
  unsigned long long ga = (unsigned long long)(size_t)g;
  asm volatile("global_load_async_to_lds_b128 %0, %1, off"
               :: "v"(lo), "v"(ga) : "memory");
}

// TDM: load a (64 x 32 f32) tile whose rows are stride_elems apart in global
// memory into LDS with a 4-dword pad after every 32 dwords (=> lda 36).
// Issued once per wave (EXEC-independent); tracked by TENSORcnt.
DEVFN void tdm_load_a(const float* gsrc, float* lbuf, unsigned stride_elems) {
  unsigned long long ga = (unsigned long long)(size_t)gsrc;
  u32x4 g0;
  g0.x = 1u;                                 // count=1, user descriptor
  g0.y = (unsigned)(size_t)lbuf;             // lds_addr (bytes)
  g0.z = (unsigned)ga;                       // global_addr[31:0]
  g0.w = (unsigned)(ga >> 32) | (2u << 30);  // global_addr[56:32] | type=2
  u32x8 g1;
  g1.s0 = 0x07120000u;        // data_size=4B, pad_enable, 32dw interval, 4dw pad
  g1.s1 = 0u;                 // tensor_dim0 = 0x40000000 (low16)
  g1.s2 = 0x4000u;            // tensor_dim0 hi16 | tensor_dim1 low16
  g1.s3 = 0x4000u | (32u << 16);  // tensor_dim1 hi16 | tile_dim0 = 32
  g1.s4 = 64u;                // tile_dim1 = 64, tile_dim2 = 0
  g1.s5 = stride_elems;       // tensor_dim0_stride[31:0]
  g1.s6 = 0u;
  g1.s7 = 0u;
  asm volatile("tensor_load_to_lds %0, %1" :: "s"(g0), "s"(g1) : "memory");
}

// ---------------------------------------------------------------------------
__global__ void k_zero(float* __restrict__ p, size_t n) {
  size_t i = (size_t)blockIdx.x * blockDim.x + threadIdx.x;
  size_t s = (size_t)gridDim.x * blockDim.x;
  for (; i < n; i += s) p[i] = 0.f;
}

// x(B,F,C,H,W) -> xp(N,F,C) = xr + pos_enc   (LDS-tiled transpose)
__global__ __launch_bounds__(256)
void k_transpose(const float* __restrict__ x, const float* __restrict__ pos,
                 float* __restrict__ xp) {
  __shared__ float tile[32][33];
  int blk = blockIdx.x;
  int hwt = blk & 31; blk >>= 5;
  int ct  = blk & 7;  blk >>= 3;
  int f   = blk & 15; int b = blk >> 4;
  int tx = threadIdx.x & 31;
  int ty = threadIdx.x >> 5;
  const float* src = x + (((size_t)(b * 16 + f) * 256 + ct * 32) * 1024) + hwt * 32;
  for (int i = 0; i < 4; ++i) {
    int cc = ty + i * 8;
    tile[cc][tx] = src[(size_t)cc * 1024 + tx];
  }
  __syncthreads();
  for (int i = 0; i < 4; ++i) {
    int hw = ty + i * 8;
    int n = b * 1024 + hwt * 32 + hw;
    xp[((size_t)n * 16 + f) * 256 + ct * 32 + tx] =
        tile[tx][hw] + pos[f * 256 + ct * 32 + tx];
  }
}

// Fused temporal attention, one workgroup (8 waves) per sequence n.
__global__ __launch_bounds__(256)
void k_attention(const float* __restrict__ xp_g, const float* __restrict__ pos,
                 const float* __restrict__ Wq, const float* __restrict__ bq,
                 const float* __restrict__ Wk, const float* __restrict__ bk,
                 const float* __restrict__ Wv, const float* __restrict__ bv,
                 const float* __restrict__ Wo, const float* __restrict__ bo,
                 const float* __restrict__ ln_g, const float* __restrict__ ln_b,
                 float* __restrict__ attn_agg) {
  constexpr int LD = 260;  // de-conflict LDS banks for column reads
  __shared__ float xp_s[16 * LD];
  __shared__ float q_s[16 * LD];
  __shared__ float k_s[16 * LD];
  __shared__ float v_s[16 * LD];
  __shared__ float at_s[16 * LD];
  __shared__ float sc_s[8][16][17];
  int n = blockIdx.x;
  int tid = threadIdx.x, lane = tid & 31, wid = tid >> 5;

  for (int i = tid; i < 16 * 256; i += 256) {
    int f = i >> 8, cc = i & 255;
    xp_s[f * LD + cc] = xp_g[(size_t)n * 4096 + i];
  }
  __syncthreads();

  // Q/K/V projections: 48 (matrix, col-tile) tasks over 8 waves
  for (int task = wid; task < 48; task += 8) {
    int mat = task >> 4, t = task & 15;
    int n0 = t * 16;
    const float* Wp = (mat == 0) ? Wq : (mat == 1) ? Wk : Wv;
    const float* bp = (mat == 0) ? bq : (mat == 1) ? bk : bv;
    float* dst = (mat == 0) ? q_s : (mat == 1) ? k_s : v_s;
    v8f acc = {};
    for (int k0 = 0; k0 < 256; k0 += 4) {
      v2f a = frag_a(xp_s, LD, lane, k0);
      v2f b = frag_b(Wp + n0, 256, lane, k0);
      acc = wmma_f32(a, b, acc);
    }
    int col = n0 + (lane & 15);
    float bias = bp[col];
    for (int r = 0; r < 8; ++r) {
      int m = r + ((lane >> 4) << 3);
      dst[m * LD + col] = acc[r] + bias;
    }
  }
  __syncthreads();

  // scores = Q_h K_h^T / sqrt(hd), wave = head
  {
    int h = wid, c0 = h * 32;
    v8f acc = {};
    for (int k0 = 0; k0 < 32; k0 += 4) {
      v2f a = frag_a(q_s + c0, LD, lane, k0);
      v2f b;  // B[k][g] = K[g][c0+k] : transposed read from K rows
      const float* kp = k_s + (size_t)(lane & 15) * LD + c0 + k0 + ((lane >> 4) << 1);
      b.x = kp[0]; b.y = kp[1];
      acc = wmma_f32(a, b, acc);
    }
    int g = lane & 15;
    for (int r = 0; r < 8; ++r) {
      int m = r + ((lane >> 4) << 3);
      sc_s[h][m][g] = acc[r] * 0.1767766953f;  // 1/sqrt(32)
    }
  }
  __syncthreads();
  if (lane < 16) {  // softmax: lane = row f, wave = head
    int h = wid, f = lane;
    float mx = -1e30f;
    for (int g = 0; g < 16; ++g) mx = fmaxf(mx, sc_s[h][f][g]);
    float s = 0.f;
    for (int g = 0; g < 16; ++g) {
      float e = __expf(sc_s[h][f][g] - mx);
      sc_s[h][f][g] = e; s += e;
    }
    float inv = 1.f / s;
    for (int g = 0; g < 16; ++g) sc_s[h][f][g] *= inv;
  }
  __syncthreads();

  // attended = attn @ V_h
  {
    int h = wid;
    for (int nt = 0; nt < 2; ++nt) {
      int c0 = h * 32 + nt * 16;
      v8f acc = {};
      for (int k0 = 0; k0 < 16; k0 += 4) {
        v2f a = frag_a(&sc_s[h][0][0], 17, lane, k0);
        v2f b = frag_b(v_s + c0, LD, lane, k0);
        acc = wmma_f32(a, b, acc);
      }
      int col = c0 + (lane & 15);
      for (int r = 0; r < 8; ++r) {
        int m = r + ((lane >> 4) << 3);
        at_s[m * LD + col] = acc[r];
      }
    }
  }
  __syncthreads();

  // out-projection + residual (xr = xp - pos_enc); result reuses q_s
  for (int t = wid; t < 16; t += 8) {
    int n0 = t * 16;
    v8f acc = {};
    for (int k0 = 0; k0 < 256; k0 += 4) {
      v2f a = frag_a(at_s, LD, lane, k0);
      v2f b = frag_b(Wo + n0, 256, lane, k0);
      acc = wmma_f32(a, b, acc);
    }
    int col = n0 + (lane & 15);
    float bias = bo[col];
    for (int r = 0; r < 8; ++r) {
      int m = r + ((lane >> 4) << 3);
      float resid = xp_s[m * LD + col] - pos[m * 256 + col];
      q_s[m * LD + col] = acc[r] + bias + resid;
    }
  }
  __syncthreads();

  // layernorm per f-row
  for (int f = wid; f < 16; f += 8) {
    float s = 0.f, s2 = 0.f;
    for (int cc = lane; cc < 256; cc += 32) {
      float v = q_s[f * LD + cc]; s += v; s2 += v * v;
    }
    for (int m = 16; m >= 1; m >>= 1) {
      s  += __shfl_xor(s, m, 32);
      s2 += __shfl_xor(s2, m, 32);
    }
    float mean = s * (1.f / 256.f);
    float var = s2 * (1.f / 256.f) - mean * mean;
    float rstd = rsqrtf(var + EPSV);
    for (int cc = lane; cc < 256; cc += 32)
      q_s[f * LD + cc] = (q_s[f * LD + cc] - mean) * rstd * ln_g[cc] + ln_b[cc];
  }
  __syncthreads();

  // temporal mean -> attn_agg(B,C,H,W)
  int bb = n >> 10, hw = n & 1023;
  for (int cc = tid; cc < 256; cc += 256) {
    float s = 0.f;
    for (int f = 0; f < 16; ++f) s += q_s[f * LD + cc];
    attn_agg[((size_t)(bb * 256 + cc)) * 1024 + hw] = s * (1.f / 16.f);
  }
}

// ConvLSTM step: implicit GEMM (OC=512) x (K = 384*9), TDM-staged weights
__global__ __launch_bounds__(512)
void k_lstm_gemm(const float* __restrict__ x, const float* __restrict__ h_prev,
                 const float* __restrict__ Wl, const float* __restrict__ bl,
                 float* __restrict__ cc, int t) {
  constexpr int KTOT = 384 * 9, KC = 32, NIT = KTOT / KC;
  __shared__ float As[2][64 * LDA];
  __shared__ float Bs[KC * 68];
  int blk = blockIdx.x;
  int pxt = blk & 15; blk >>= 4;
  int oct = blk & 7;  int b = blk >> 3;
  int oc0 = oct * 64;
  int y0 = (pxt >> 2) * 8, x0 = (pxt & 3) * 8;
  int tid = threadIdx.x, lane = tid & 31, wid = tid >> 5;
  int wm = wid >> 2, wn = wid & 3;
  const float* Abase = Wl + (size_t)oc0 * KTOT;

  if (wid == 0) tdm_load_a(Abase, As[0], KTOT);  // preload tile 0
  v8f acc = {};
  for (int it = 0; it < NIT; ++it) {
    int k0 = it * KC;
    for (int i = tid; i < KC * 64; i += 512) {  // im2col gather (zero-padded)
      int k = i >> 6, p = i & 63;
      int kk = k0 + k;
      int ic = kk / 9, off = kk % 9;
      int ky = off / 3, kx = off % 3;
      int iy = y0 + (p >> 3) + ky - 1;
      int ix = x0 + (p & 7) + kx - 1;
      float v = 0.f;
      if (iy >= 0 && iy < 32 && ix >= 0 && ix < 32) {
        if (ic < 256)
          v = x[(((size_t)(b * 16 + t) * 256 + ic) * 1024) + iy * 32 + ix];
        else
          v = h_prev[(((size_t)b * 128 + (ic - 256)) * 1024) + iy * 32 + ix];
      }
      Bs[k * 68 + p] = v;
    }
    if (wid == 0) {
      if (it + 1 < NIT) {
        tdm_load_a(Abase + k0 + KC, As[(it + 1) & 1], KTOT);
        __builtin_amdgcn_s_wait_tensorcnt(1);  // tile `it` complete (in-order)
      } else {
        __builtin_amdgcn_s_wait_tensorcnt(0);
      }
    }
    __syncthreads();
    const float* Ap = As[it & 1] + wm * 16 * LDA;
    const float* Bp = Bs + wn * 16;
    for (int kk = 0; kk < KC; kk += 4) {
      v2f a = frag_a(Ap, LDA, lane, kk);
      v2f b2 = frag_b(Bp, 68, lane, kk);
      acc = wmma_f32(a, b2, acc);
    }
    __syncthreads();
  }
  int col = wn * 16 + (lane & 15);
  int py = y0 + (col >> 3), px = x0 + (col & 7);
  for (int r = 0; r < 8; ++r) {
    int m = r + ((lane >> 4) << 3);
    int o = oc0 + wm * 16 + m;
    cc[(((size_t)b * 512 + o) * 1024) + py * 32 + px] = acc[r] + bl[o];
  }
}

__global__ void k_lstm_gate(const float* __restrict__ cc, float* __restrict__ h,
                            float* __restrict__ c, float* __restrict__ acc) {
  int i = blockIdx.x * blockDim.x + threadIdx.x;
  if (i >= B_ * Hd_ * HW) return;
  int b = i >> 17;
  int rem = i & 131071;
  size_t base = (size_t)b * 512 * 1024;
  float iv = cc[base + rem];
  float fv = cc[base + 131072 + rem];
  float gv = cc[base + 262144 + rem];
  float ov = cc[base + 393216 + rem];
  float si = 1.f / (1.f + __expf(-iv));
  float sf = 1.f / (1.f + __expf(-fv));
  float so = 1.f / (1.f + __expf(-ov));
  float cn = sf * c[i] + si * tanhf(gv);
  float hn = so * tanhf(cn);
  c[i] = cn; h[i] = hn;
  acc[i] += hn * (1.f / 16.f);  // temporal mean folded in (async-copy friendly)
}

// Conv3D implicit GEMM, TDM-staged weights. pass 1: BN stats. pass 2:
// recompute, normalize, ReLU, temporal mean (avoids 34MB intermediate).
__global__ __launch_bounds__(512)
void k_conv3d(const float* __restrict__ x, const float* __restrict__ W3,
              const float* __restrict__ b3, const float* __restrict__ bn_g,
              const float* __restrict__ bn_b, const float* __restrict__ stats,
              float* __restrict__ out, int pass) {
  constexpr int KTOT = 256 * 27, KC = 32, NIT = KTOT / KC;
  __shared__ float As[2][64 * LDA];
  __shared__ float Bs[KC * 68];
  __shared__ float red[128];
  int blk = blockIdx.x;
  int pxt = blk & 15; blk >>= 4;
  int oct = blk & 1;  blk >>= 1;
  int f = blk & 15;   int b = blk >> 4;
  int oc0 = oct * 64;
  int y0 = (pxt >> 2) * 8, x0 = (pxt & 3) * 8;
  int tid = threadIdx.x, lane = tid & 31, wid = tid >> 5;
  int wm = wid >> 2, wn = wid & 3;
  const float* Abase = W3 + (size_t)oc0 * KTOT;

  if (wid == 0) tdm_load_a(Abase, As[0], KTOT);
  v8f acc = {};
  for (int it = 0; it < NIT; ++it) {
    int k0 = it * KC;
    for (int i = tid; i < KC * 64; i += 512) {
      int k = i >> 6, p = i & 63;
      int kk = k0 + k;
      int ic = kk / 27, off = kk % 27;
      int kf = off / 9, r9 = off % 9;
      int ky = r9 / 3, kx = r9 % 3;
      int fi = f + kf - 1;
      int iy = y0 + (p >> 3) + ky - 1;
      int ix = x0 + (p & 7) + kx - 1;
      float v = 0.f;
      if (fi >= 0 && fi < 16 && iy >= 0 && iy < 32 && ix >= 0 && ix < 32)
        v = x[(((size_t)(b * 16 + fi) * 256 + ic) * 1024) + iy * 32 + ix];
      Bs[k * 68 + p] = v;
    }
    if (wid == 0) {
      if (it + 1 < NIT) {
        tdm_load_a(Abase + k0 + KC, As[(it + 1) & 1], KTOT);
        __builtin_amdgcn_s_wait_tensorcnt(1);
      } else {
        __builtin_amdgcn_s_wait_tensorcnt(0);
      }
    }
    __syncthreads();
    const float* Ap = As[it & 1] + wm * 16 * LDA;
    const float* Bp = Bs + wn * 16;
    for (int kk = 0; kk < KC; kk += 4) {
      v2f a = frag_a(Ap, LDA, lane, kk);
      v2f b2 = frag_b(Bp, 68, lane, kk);
      acc = wmma_f32(a, b2, acc);
    }
    __syncthreads();
  }
  int col = wn * 16 + (lane & 15);
  int py = y0 + (col >> 3), px = x0 + (col & 7);
  if (pass == 1) {
    if (tid < 128) red[tid] = 0.f;
    __syncthreads();
    for (int r = 0; r < 8; ++r) {
      int m = r + ((lane >> 4) << 3);
      int ch = oc0 + wm * 16 + m;
      float y = acc[r] + b3[ch];
      float s = y, s2 = y * y;
      for (int msk = 1; msk < 16; msk <<= 1) {
        s  += __shfl_xor(s, msk, 16);
        s2 += __shfl_xor(s2, msk, 16);
      }
      if ((lane & 15) == 0) {
        atomicAdd(&red[wm * 16 + m], s);
        atomicAdd(&red[64 + wm * 16 + m], s2);
      }
    }
    __syncthreads();
    if (tid < 64) atomicAdd(&out[oc0 + tid], red[tid]);
    else if (tid < 128) atomicAdd(&out[128 + oc0 + tid - 64], red[tid]);
  } else {
    for (int r = 0; r < 8; ++r) {
      int m = r + ((lane >> 4) << 3);
      int ch = oc0 + wm * 16 + m;
      float y = acc[r] + b3[ch];
      float v = fmaxf(0.f, (y - stats[256 + ch]) * stats[384 + ch] * bn_g[ch] + bn_b[ch]);
      atomicAdd(&out[(((size_t)b * 128 + ch) * 1024) + py * 32 + px], v * (1.f / 16.f));
    }
  }
}

__global__ void k_bn_stats(float* __restrict__ stats, int nch, float invn) {
  int c = blockIdx.x * blockDim.x + threadIdx.x;
  if (c < nch) {
    float m = stats[c] * invn;
    float v = stats[nch + c] * invn - m * m;
    stats[2 * nch + c] = m;
    stats[3 * nch + c] = rsqrtf(v + EPSV);
  }
}

// Fusion 1x1 conv: (256 x 512) @ (512 x 4096) via WMMA; A and B staged with
// per-lane async global->LDS b128 copies (ASYNCcnt), A double-buffered.
__global__ __launch_bounds__(512)
void k_fusion(const float* __restrict__ attn_agg, const float* __restrict__ lstm_acc,
              const float* __restrict__ c3a, const float* __restrict__ Wf,
              const float* __restrict__ bf, float* __restrict__ out,
              float* __restrict__ stats2) {
  constexpr int KTOT = 512, KC = 32, NIT = KTOT / KC;
  __shared__ float As[2][64 * LDA];
  __shared__ float Bs[KC * 68];
  __shared__ float red[128];
  int blk = blockIdx.x;
  int pxt = blk & 63; int oct = blk >> 6;
  int oc0 = oct * 64;
  int p0g = pxt * 64;
  int tid = threadIdx.x, lane = tid & 31, wid = tid >> 5;
  int wm = wid >> 2, wn = wid & 3;
  int b4 = p0g >> 10, hw0 = p0g & 1023;  // tile lies within one batch image

  int ar = tid >> 3, acg = (tid & 7) << 2;   // A: row, 4-dword col group
  int bk = tid >> 4, bcg = (tid & 15) << 2;  // B: k row, 4-dword col group
  const float* Abase = Wf + (size_t)(oc0 + ar) * KTOT + acg;

  async_b128(&As[0][ar * LDA + acg], Abase);  // preload A tile 0
  v8f acc = {};
  for (int it = 0; it < NIT; ++it) {
    int k0 = it * KC;
    int k = k0 + bk;
    const float* src;
    if (k < 256)      src = attn_agg + ((size_t)(b4 * 256 + k)) * 1024 + hw0;
    else if (k < 384) src = lstm_acc + ((size_t)(b4 * 128 + (k - 256))) * 1024 + hw0;
    else              src = c3a + ((size_t)(b4 * 128 + (k - 384))) * 1024 + hw0;
    async_b128(&Bs[bk * 68 + bcg], src + bcg);
    if (it + 1 < NIT) {
      async_b128(&As[(it + 1) & 1][ar * LDA + acg], Abase + k0 + KC);
      asm volatile("s_wait_asynccnt 0x1" ::: "memory");  // B(it)+A(it) done
    } else {
      asm volatile("s_wait_asynccnt 0x0" ::: "memory");
    }
    __syncthreads();
    const float* Ap = As[it & 1] + wm * 16 * LDA;
    const float* Bp = Bs + wn * 16;
    for (int kk = 0; kk < KC; kk += 4) {
      v2f a = frag_a(Ap, LDA, lane, kk);
      v2f b2 = frag_b(Bp, 68, lane, kk);
      acc = wmma_f32(a, b2, acc);
    }
    __syncthreads();
  }
  if (tid < 128) red[tid] = 0.f;
  __syncthreads();
  int p = p0g + wn * 16 + (lane & 15);
  int b = p >> 10, hw = p & 1023;
  for (int r = 0; r < 8; ++r) {
    int m = r + ((lane >> 4) << 3);
    int ch = oc0 + wm * 16 + m;
    float y = acc[r] + bf[ch];
    out[((size_t)(b * 256 + ch)) * 1024 + hw] = y;
    float s = y, s2 = y * y;
    for (int msk = 1; msk < 16; msk <<= 1) {
      s  += __shfl_xor(s, msk, 16);
      s2 += __shfl_xor(s2, msk, 16);
    }
    if ((lane & 15) == 0) {
      atomicAdd(&red[wm * 16 + m], s);
      atomicAdd(&red[64 + wm * 16 + m], s2);
    }
  }
  __syncthreads();
  if (tid < 64) atomicAdd(&stats2[oc0 + tid], red[tid]);
  else if (tid < 128) atomicAdd(&stats2[256 + oc0 + tid - 64], red[tid]);
}

__global__ void k_bn2d_apply(float* __restrict__ out, const float* __restrict__ stats2,
                             const float* __restrict__ g, const float* __restrict__ bb) {
  int i = blockIdx.x * blockDim.x + threadIdx.x;
  if (i >= B_ * C_ * HW) return;
  int ch = (i >> 10) & 255;
  out[i] = (out[i] - stats2[512 + ch]) * stats2[768 + ch] * g[ch] + bb[ch];
}

// ---------------------------------------------------------------------------
extern "C" void kernel_launch(void* const* d_in, const int* in_sizes, int n_in,
                              void* d_out, int out_size, void* d_ws, size_t ws_size,
                              hipStream_t stream) {
  const float* x      = (const float*)d_in[0];
  const float* pos    = (const float*)d_in[1];
  const float* Wq     = (const float*)d_in[2];
  const float* bq     = (const float*)d_in[3];
  const float* Wk     = (const float*)d_in[4];
  const float* bk     = (const float*)d_in[5];
  const float* Wv     = (const float*)d_in[6];
  const float* bv     = (const float*)d_in[7];
  const float* Wo     = (const float*)d_in[8];
  const float* bo     = (const float*)d_in[9];
  const float* ln_g   = (const float*)d_in[10];
  const float* ln_b   = (const float*)d_in[11];
  const float* lstm_W = (const float*)d_in[12];
  const float* lstm_b = (const float*)d_in[13];
  const float* c3d_W  = (const float*)d_in[14];
  const float* c3d_b  = (const float*)d_in[15];
  const float* bn3d_g = (const float*)d_in[16];
  const float* bn3d_b = (const float*)d_in[17];
  const float* fus_W  = (const float*)d_in[18];
  const float* fus_b  = (const float*)d_in[19];
  const float* bn2d_g = (const float*)d_in[20];
  const float* bn2d_b = (const float*)d_in[21];

  float* ws = (float*)d_ws;
  float* xp       = ws + OFF_XP;
  float* attn_agg = ws + OFF_ATTN;
  float* h        = ws + OFF_H;
  float* c        = ws + OFF_C;
  float* acc      = ws + OFF_ACC;
  float* c3a      = ws + OFF_C3A;
  float* s3       = ws + OFF_S3;
  float* s2       = ws + OFF_S2;
  float* cc       = ws + OFF_CC;
  float* out      = (float*)d_out;

  // zero: h, c, acc, conv3d_agg, stat buffers (contiguous range)
  size_t nz = (OFF_S2 + 1024) - OFF_H;
  k_zero<<<1024, 256, 0, stream>>>(ws + OFF_H, nz);

  k_transpose<<<16384, 256, 0, stream>>>(x, pos, xp);
  k_attention<<<4096, 256, 0, stream>>>(xp, pos, Wq, bq, Wk, bk, Wv, bv,
                                        Wo, bo, ln_g, ln_b, attn_agg);
  for (int t = 0; t < 16; ++t) {
    k_lstm_gemm<<<512, 512, 0, stream>>>(x, h, lstm_W, lstm_b, cc, t);
    k_lstm_gate<<<2048, 256, 0, stream>>>(cc, h, c, acc);
  }
  k_conv3d<<<2048, 512, 0, stream>>>(x, c3d_W, c3d_b, bn3d_g, bn3d_b, s3, s3, 1);
  k_bn_stats<<<1, 128, 0, stream>>>(s3, 128, 1.f / 65536.f);
  k_conv3d<<<2048, 512, 0, stream>>>(x, c3d_W, c3d_b, bn3d_g, bn3d_b, s3, c3a, 2);
  k_fusion<<<256, 512, 0, stream>>>(attn_agg, acc, c3a, fus_W, fus_b, out, s2);
  k_bn_stats<<<1, 256, 0, stream>>>(s2, 256, 1.f / 4096.f);
  k_bn2d_apply<<<4096, 256, 0, stream>>>(out, s2, bn2d_g, bn2d_b);
}